// GaussianAdapter_57294863729282
// MI455X (gfx1250) — compile-verified
//
#include <hip/hip_runtime.h>
#include <cstdint>

// ---- problem dims (from reference module constants) ----
#define B_DIM   4
#define V_DIM   4
#define H_DIM   256
#define W_DIM   384
#define HW      (H_DIM * W_DIM)          // 98304 pixels per view
#define NVIEW   (B_DIM * V_DIM)          // 16
#define NPIX    ((long)NVIEW * (long)HW) // 1572864 total pixels

#define SCALE_MIN_F 1e-05f
#define SCALE_MAX_F 30.0f
#define EPS_Q       1e-08f

#define TPB             256
#define TILES_PER_BLOCK 4

#if defined(__gfx1250__)
#define USE_ASYNC 1
#else
#define USE_ASYNC 0
#endif

// 12-byte vector for b96-mergeable stores (align 4, packed)
struct __align__(4) F3 { float x, y, z; };

// per-view constant block layout in d_ws (32 floats per view):
// [0..8]  Rc2w row-major
// [9..11] origin
// [12..20] Kinv row-major (inverse of normalized intrinsics)
// [21..24] c2w quaternion (w,x,y,z)
// [25]    mult scalar
#define CS_STRIDE 32

// ---------------- setup kernel: per-view constants ----------------
__global__ void gs_setup(const float* __restrict__ ext,
                         const float* __restrict__ intr,
                         float* __restrict__ cs) {
    const int v = threadIdx.x;
    if (v >= NVIEW) return;

    const float* E = ext + v * 16;        // 4x4 row-major
    // Rc2w = transpose(Rw2c)
    float R[3][3];
#pragma unroll
    for (int r = 0; r < 3; ++r)
#pragma unroll
        for (int c = 0; c < 3; ++c)
            R[r][c] = E[c * 4 + r];
    const float t0 = E[3], t1 = E[7], t2 = E[11];
    const float o0 = -(R[0][0]*t0 + R[0][1]*t1 + R[0][2]*t2);
    const float o1 = -(R[1][0]*t0 + R[1][1]*t1 + R[1][2]*t2);
    const float o2 = -(R[2][0]*t0 + R[2][1]*t1 + R[2][2]*t2);

    // normalized intrinsics: row0 /= W, row1 /= H, row2 *= 1
    const float* I = intr + v * 9;
    const float iw = 1.0f / (float)W_DIM, ih = 1.0f / (float)H_DIM;
    float K[3][3];
    K[0][0] = I[0]*iw; K[0][1] = I[1]*iw; K[0][2] = I[2]*iw;
    K[1][0] = I[3]*ih; K[1][1] = I[4]*ih; K[1][2] = I[5]*ih;
    K[2][0] = I[6];    K[2][1] = I[7];    K[2][2] = I[8];

    // full 3x3 inverse (adjugate)
    const float c00 =  (K[1][1]*K[2][2] - K[1][2]*K[2][1]);
    const float c01 = -(K[1][0]*K[2][2] - K[1][2]*K[2][0]);
    const float c02 =  (K[1][0]*K[2][1] - K[1][1]*K[2][0]);
    const float det = K[0][0]*c00 + K[0][1]*c01 + K[0][2]*c02;
    const float id  = 1.0f / det;
    float Ki[3][3];
    Ki[0][0] = c00*id;
    Ki[0][1] = (K[0][2]*K[2][1] - K[0][1]*K[2][2])*id;
    Ki[0][2] = (K[0][1]*K[1][2] - K[0][2]*K[1][1])*id;
    Ki[1][0] = c01*id;
    Ki[1][1] = (K[0][0]*K[2][2] - K[0][2]*K[2][0])*id;
    Ki[1][2] = (K[0][2]*K[1][0] - K[0][0]*K[1][2])*id;
    Ki[2][0] = c02*id;
    Ki[2][1] = (K[0][1]*K[2][0] - K[0][0]*K[2][1])*id;
    Ki[2][2] = (K[0][0]*K[1][1] - K[0][1]*K[1][0])*id;

    // mult = 0.1 * sum_i sum_j inv2[i][j] * pixel_size[j]
    const float a = K[0][0], b = K[0][1], c = K[1][0], d = K[1][1];
    const float det2 = a*d - b*c;
    const float mult = 0.1f * ((d - c)*iw + (a - b)*ih) / det2;

    // _mat_to_quat_wxyz(Rc2w)
    const float m00=R[0][0], m01=R[0][1], m02=R[0][2];
    const float m10=R[1][0], m11=R[1][1], m12=R[1][2];
    const float m20=R[2][0], m21=R[2][1], m22=R[2][2];
    float qa[4];
    qa[0] = sqrtf(fmaxf(0.0f, 1.0f + m00 + m11 + m22));
    qa[1] = sqrtf(fmaxf(0.0f, 1.0f + m00 - m11 - m22));
    qa[2] = sqrtf(fmaxf(0.0f, 1.0f - m00 + m11 - m22));
    qa[3] = sqrtf(fmaxf(0.0f, 1.0f - m00 - m11 + m22));
    const float cand[4][4] = {
        { qa[0]*qa[0], m21 - m12,  m02 - m20,  m10 - m01 },
        { m21 - m12,   qa[1]*qa[1], m10 + m01, m02 + m20 },
        { m02 - m20,   m10 + m01,  qa[2]*qa[2], m12 + m21 },
        { m10 - m01,   m20 + m02,  m21 + m12,  qa[3]*qa[3] }};
    int idx = 0; float best = qa[0];
#pragma unroll
    for (int i = 1; i < 4; ++i) { if (qa[i] > best) { best = qa[i]; idx = i; } }
    const float sdiv = 1.0f / (2.0f * fmaxf(qa[idx], 0.1f));
    float q0 = cand[idx][0]*sdiv, q1 = cand[idx][1]*sdiv,
          q2 = cand[idx][2]*sdiv, q3 = cand[idx][3]*sdiv;
    const float qin = 1.0f / sqrtf(q0*q0 + q1*q1 + q2*q2 + q3*q3);
    q0 *= qin; q1 *= qin; q2 *= qin; q3 *= qin;

    float* C = cs + v * CS_STRIDE;
    C[0]=R[0][0]; C[1]=R[0][1]; C[2]=R[0][2];
    C[3]=R[1][0]; C[4]=R[1][1]; C[5]=R[1][2];
    C[6]=R[2][0]; C[7]=R[2][1]; C[8]=R[2][2];
    C[9]=o0; C[10]=o1; C[11]=o2;
    C[12]=Ki[0][0]; C[13]=Ki[0][1]; C[14]=Ki[0][2];
    C[15]=Ki[1][0]; C[16]=Ki[1][1]; C[17]=Ki[1][2];
    C[18]=Ki[2][0]; C[19]=Ki[2][1]; C[20]=Ki[2][2];
    C[21]=q0; C[22]=q1; C[23]=q2; C[24]=q3;
    C[25]=mult;
}

// ---------------- main per-pixel kernel ----------------
__global__ __launch_bounds__(TPB) void gs_main(
    const float* __restrict__ depths, const float* __restrict__ opacs,
    const float* __restrict__ raw, const float* __restrict__ cs,
    float* __restrict__ out) {
    const int tid  = threadIdx.x;
    const int view = blockIdx.y;
    const long vbase = (long)view * (long)HW;

    // per-view constants (uniform per block)
    const float* C = cs + view * CS_STRIDE;
    const float R00=C[0], R01=C[1], R02=C[2];
    const float R10=C[3], R11=C[4], R12=C[5];
    const float R20=C[6], R21=C[7], R22=C[8];
    const float ox=C[9], oy=C[10], oz=C[11];
    const float K00=C[12], K01=C[13], K02=C[14];
    const float K10=C[15], K11=C[16], K12=C[17];
    const float K20=C[18], K21=C[19], K22=C[20];
    const float aw=C[21], ax=C[22], ay=C[23], az=C[24];
    const float mult=C[25];

    float* out_means  = out;
    float* out_scales = out + NPIX * 3;
    float* out_rot    = out + NPIX * 6;
    float* out_harm   = out + NPIX * 10;
    float* out_opac   = out + NPIX * 13;

    const float iw = 1.0f / (float)W_DIM, ih = 1.0f / (float)H_DIM;
    const int tile0 = blockIdx.x * TILES_PER_BLOCK;

#if USE_ASYNC
    __shared__ __align__(16) float lraw[2][TPB * 12];
    // one async global->LDS batch (3 x b128 = 48B/lane); offset:N advances
    // both the LDS and global addresses per the ISA pseudocode.
#define ISSUE_TILE(bufi, tidx)                                                  \
    do {                                                                        \
        const float* g_ = raw + ((vbase + (long)(tidx) * TPB + tid) * 12);      \
        uint32_t l_ = (uint32_t)(uintptr_t)(void*)&lraw[(bufi)][tid * 12];      \
        asm volatile(                                                           \
            "global_load_async_to_lds_b128 %0, %1, off\n\t"                     \
            "global_load_async_to_lds_b128 %0, %1, off offset:16\n\t"           \
            "global_load_async_to_lds_b128 %0, %1, off offset:32"               \
            :: "v"(l_), "v"(g_) : "memory");                                    \
    } while (0)

    ISSUE_TILE(0, tile0);
#endif

    for (int it = 0; it < TILES_PER_BLOCK; ++it) {
        const int  p  = (tile0 + it) * TPB + tid;   // pixel within view
        const long gp = vbase + (long)p;            // global pixel index
        float4 r0, r1, r2;
#if USE_ASYNC
        if (it + 1 < TILES_PER_BLOCK) {
            ISSUE_TILE((it + 1) & 1, tile0 + it + 1);
            asm volatile("s_wait_asynccnt 0x3" ::: "memory");
        } else {
            asm volatile("s_wait_asynccnt 0x0" ::: "memory");
        }
        const float* lr = &lraw[it & 1][tid * 12];
        r0 = *(const float4*)(lr + 0);
        r1 = *(const float4*)(lr + 4);
        r2 = *(const float4*)(lr + 8);
#else
        const float* gr = raw + gp * 12;
        r0 = *(const float4*)(gr + 0);
        r1 = *(const float4*)(gr + 4);
        r2 = *(const float4*)(gr + 8);
#endif
        const float depth = depths[gp];
        const float op    = opacs[gp];

        const int w = p % W_DIM;
        const int h = p / W_DIM;

        // ---- means ----
        const float x = ((float)w + 0.5f) * iw + r0.x * iw;
        const float y = ((float)h + 0.5f) * ih + r0.y * ih;
        float dc0 = K00 * x + K01 * y + K02;
        float dc1 = K10 * x + K11 * y + K12;
        float dc2 = K20 * x + K21 * y + K22;
        const float invn = 1.0f / sqrtf(dc0*dc0 + dc1*dc1 + dc2*dc2);
        dc0 *= invn; dc1 *= invn; dc2 *= invn;
        const float dw0 = R00*dc0 + R01*dc1 + R02*dc2;
        const float dw1 = R10*dc0 + R11*dc1 + R12*dc2;
        const float dw2 = R20*dc0 + R21*dc1 + R22*dc2;
        F3 mv; mv.x = ox + dw0 * depth; mv.y = oy + dw1 * depth; mv.z = oz + dw2 * depth;
        *(F3*)(out_means + gp * 3) = mv;   // b96-mergeable store

        // ---- scales ----
        const float dm = depth * mult;
        const float s0 = SCALE_MIN_F + (SCALE_MAX_F - SCALE_MIN_F) / (1.0f + expf(-r0.z));
        const float s1 = SCALE_MIN_F + (SCALE_MAX_F - SCALE_MIN_F) / (1.0f + expf(-r0.w));
        const float s2 = SCALE_MIN_F + (SCALE_MAX_F - SCALE_MIN_F) / (1.0f + expf(-r1.x));
        F3 sv; sv.x = s0 * dm; sv.y = s1 * dm; sv.z = s2 * dm;
        *(F3*)(out_scales + gp * 3) = sv;

        // ---- rotations: world_q = c2w_q * cam_q (wxyz) ----
        float bx = r1.y, by = r1.z, bz = r1.w, bw = r2.x;
        const float qn = sqrtf(bx*bx + by*by + bz*bz + bw*bw);
        const float qi = 1.0f / (qn + EPS_Q);
        bw *= qi; bx *= qi; by *= qi; bz *= qi;
        float4 qv;
        qv.x = aw*bw - ax*bx - ay*by - az*bz;
        qv.y = aw*bx + ax*bw + ay*bz - az*by;
        qv.z = aw*by - ax*bz + ay*bw + az*bx;
        qv.w = aw*bz + ax*by - ay*bx + az*bw;
        *(float4*)(out_rot + gp * 4) = qv; // 16B aligned -> global_store_b128

        // ---- harmonics & opacity ----
        F3 hv; hv.x = r2.y; hv.y = r2.z; hv.z = r2.w;
        *(F3*)(out_harm + gp * 3) = hv;
        out_opac[gp] = op;
    }
}

extern "C" void kernel_launch(void* const* d_in, const int* in_sizes, int n_in,
                              void* d_out, int out_size, void* d_ws, size_t ws_size,
                              hipStream_t stream) {
    const float* ext    = (const float*)d_in[0];
    const float* intr   = (const float*)d_in[1];
    const float* depths = (const float*)d_in[2];
    const float* opacs  = (const float*)d_in[3];
    const float* raw    = (const float*)d_in[4];
    float* ws  = (float*)d_ws;
    float* out = (float*)d_out;

    gs_setup<<<dim3(1), dim3(NVIEW), 0, stream>>>(ext, intr, ws);

    dim3 grid(HW / (TPB * TILES_PER_BLOCK), NVIEW); // 96 x 16
    gs_main<<<grid, dim3(TPB), 0, stream>>>(depths, opacs, raw, ws, out);
}